// SINDyLossMixin_2379411882188
// MI455X (gfx1250) — compile-verified
//
#include <hip/hip_runtime.h>

// CDNA5 / gfx1250: fp32 WMMA RK4 kernel, spill-free, branch-free epilogues.
// D = A(16x4) x B(4x16) + C(16x16), wave32.

typedef float v2f __attribute__((ext_vector_type(2)));
typedef float v8f __attribute__((ext_vector_type(8)));

#define PITCH  132  // row-major staging pitch: 528B rows (16B aligned, conflict-free b64 frags)
#define LPITCH 66   // lane-major pitch: base = 2*lane banks -> conflict-free, 8B aligned
#define WPB    8    // waves per block

constexpr int kH = 128;
constexpr int NROWS  = 256 * 1023;      // 261888 rows of y (multiple of 16)
constexpr int TILES  = NROWS / 16;      // 16368
constexpr int BLOCKS = TILES / WPB;     // 2046 (exact)

// acc(16x128 as 8 col-tiles) = stage(16x128) @ C^T, K in slices of 4
__device__ __forceinline__ void gemm_stage(const float* stage, const float* Clds,
                                           int lo, int hi, v8f acc[8])
{
    #pragma unroll
    for (int ct = 0; ct < 8; ++ct) acc[ct] = v8f{};

    #pragma unroll 2                  // pipelines loads vs wmma w/o pressure blowup
    for (int k = 0; k < 32; ++k) {
        // A frag: (K,K+1) low half-wave / (K+2,K+3) high half-wave, one b64
        const v2f a = *(const v2f*)(stage + lo * PITCH + 4 * k + 2 * hi);
        #pragma unroll
        for (int ct = 0; ct < 8; ++ct) {
            // B frag: B[K][n] = C[n][K]; one b64 from row-major C
            const v2f bf = *(const v2f*)(Clds + (16 * ct + lo) * PITCH + 4 * k + 2 * hi);
            acc[ct] = __builtin_amdgcn_wmma_f32_16x16x4_f32(
                false, a, false, bf, (short)0, acc[ct], false, false);
        }
    }
}

__global__ __launch_bounds__(32 * WPB)
void sindy_rk4_loss(const float* __restrict__ x,
                    const float* __restrict__ coef,
                    float* __restrict__ out)
{
    __shared__ float Clds[128 * PITCH];        // C row-major, padded      (66 KB)
    __shared__ float stg[WPB][16 * PITCH];     // per-wave stage input
    __shared__ float ybuf[WPB][32 * LPITCH];   // pristine y, lane-major
    __shared__ float y1buf[WPB][32 * LPITCH];  // running y1, lane-major

    const int tid = threadIdx.x;

    // ---- stage C (128x128 f32) into LDS, row-major with padded pitch ----
    for (int i = tid; i < 128 * 32; i += 32 * WPB) {
        const int r  = i >> 5;
        const int c4 = (i & 31) << 2;
        *(float4*)&Clds[r * PITCH + c4] = *(const float4*)&coef[r * 128 + c4];
    }
    __syncthreads();

    const int wave = tid >> 5;
    const int lane = tid & 31;
    const int lo   = lane & 15;
    const int hi   = lane >> 4;
    float* stage  = stg[wave];
    float* ylane  = &ybuf[wave][lane * LPITCH];
    float* y1lane = &y1buf[wave][lane * LPITCH];

    const unsigned t = blockIdx.x * WPB + wave;   // 16-row tile index

    // ---- load y tile (16 rows) row-major into wave-private staging ----
    // row n = 16t+i -> batch b = n/1023, step s = n%1023  (y = x[:, :-1, :])
    for (int i = 0; i < 16; ++i) {
        const unsigned n = 16u * t + (unsigned)i;
        const unsigned b = n / 1023u;
        const unsigned s = n - 1023u * b;
        const float* row = x + (size_t)(b * 1024u + s) * 128u;
        *(float4*)&stage[i * PITCH + 4 * lane] = *(const float4*)&row[4 * lane];
    }

    // ---- cache this lane's C/D-layout elements of y: pristine + running y1 ----
    // element (m = v + 8*hi, col = 16*ct + lo) lives at lane index ct*8+v
    #pragma unroll
    for (int ct = 0; ct < 8; ++ct)
        #pragma unroll
        for (int v = 0; v < 8; ++v) {
            const float yv = stage[(v + 8 * hi) * PITCH + 16 * ct + lo];
            ylane [ct * 8 + v] = yv;
            y1lane[ct * 8 + v] = yv;   // y1 starts at y; gains w_s/6 * k_s per stage
        }

    // ---- stages 1..3: k = f(in); y1 += (w/6)k; next in = y + c*k ----
    #pragma unroll 1
    for (int st = 0; st < 3; ++st) {
        v8f acc[8];
        gemm_stage(stage, Clds, lo, hi, acc);

        const float w  = (st == 0) ? (1.0f / 6.0f) : (2.0f / 6.0f);  // SALU select
        const float ci = (st == 2) ? 1.0f : 0.5f;

        // (a) y1 += w*k : batched float2 RMW (loads clause, then stores)
        #pragma unroll
        for (int ct = 0; ct < 8; ++ct) {
            float2 y1p[4];
            #pragma unroll
            for (int p = 0; p < 4; ++p) y1p[p] = *(const float2*)&y1lane[ct * 8 + 2 * p];
            #pragma unroll
            for (int p = 0; p < 4; ++p) {
                y1p[p].x += w * acc[ct][2 * p];
                y1p[p].y += w * acc[ct][2 * p + 1];
            }
            #pragma unroll
            for (int p = 0; p < 4; ++p) *(float2*)&y1lane[ct * 8 + 2 * p] = y1p[p];
        }

        // (b) next stage input = y + ci*k : scatter to row-major staging
        #pragma unroll
        for (int ct = 0; ct < 8; ++ct) {
            float2 yp[4];
            #pragma unroll
            for (int p = 0; p < 4; ++p) yp[p] = *(const float2*)&ylane[ct * 8 + 2 * p];
            #pragma unroll
            for (int v = 0; v < 8; ++v) {
                const float ye = (v & 1) ? yp[v >> 1].y : yp[v >> 1].x;
                stage[(v + 8 * hi) * PITCH + 16 * ct + lo] = ye + ci * acc[ct][v];
            }
        }
    }

    // ---- stage 4 fused with loss: y1_final = y1 + k4/6 (read-only, no store) ----
    v8f acc[8];
    gemm_stage(stage, Clds, lo, hi, acc);

    float lsum = 0.0f;
    #pragma unroll
    for (int v = 0; v < 8; ++v) {
        const unsigned n = 16u * t + (unsigned)(v + 8 * hi);
        const unsigned b = n / 1023u;
        const unsigned s = n - 1023u * b;
        const float* x1row = x + (size_t)(b * 1024u + s + 1u) * 128u;
        #pragma unroll
        for (int ct = 0; ct < 8; ++ct) {
            const float y1 = y1lane[ct * 8 + v] + (1.0f / 6.0f) * acc[ct][v];
            const float d  = y1 - x1row[16 * ct + lo];
            lsum += d * d;
        }
    }

    // wave32 reduction + scaled atomic accumulate
    #pragma unroll
    for (int off = 16; off > 0; off >>= 1)
        lsum += __shfl_xor(lsum, off, 32);
    if (lane == 0)
        atomicAdd(out, lsum * (1.0f / ((float)NROWS * (float)kH)));
}

// 0.001 * mean((C*C)^2) over 128x128
__global__ __launch_bounds__(256)
void l2_kernel(const float* __restrict__ coef, float* __restrict__ out)
{
    float a = 0.0f;
    for (int i = threadIdx.x; i < 128 * 128; i += 256) {
        const float e = coef[i] * coef[i];
        a += e * e;
    }
    #pragma unroll
    for (int off = 16; off > 0; off >>= 1)
        a += __shfl_xor(a, off, 32);

    __shared__ float ws[8];
    if ((threadIdx.x & 31) == 0) ws[threadIdx.x >> 5] = a;
    __syncthreads();
    if (threadIdx.x == 0) {
        float s = 0.0f;
        #pragma unroll
        for (int i = 0; i < 8; ++i) s += ws[i];
        atomicAdd(out, s * (0.001f / 16384.0f));
    }
}

extern "C" void kernel_launch(void* const* d_in, const int* in_sizes, int n_in,
                              void* d_out, int out_size, void* d_ws, size_t ws_size,
                              hipStream_t stream)
{
    const float* x    = (const float*)d_in[0];
    const float* coef = (const float*)d_in[1];
    float* out        = (float*)d_out;

    hipMemsetAsync(out, 0, sizeof(float), stream);      // graph-capture safe
    l2_kernel<<<1, 256, 0, stream>>>(coef, out);
    sindy_rk4_loss<<<BLOCKS, 32 * WPB, 0, stream>>>(x, coef, out);
}